// AnimNeRF_51908974739808
// MI455X (gfx1250) — compile-verified
//
#include <hip/hip_runtime.h>
#include <math.h>

// CDNA5 / gfx1250, wave32.
// Distance tile (16 verts x 16 points) per V_WMMA_F32_16X16X4_F32 with
// A = (vx,vy,vz,|v|^2), B = (-2x,-2y,-2z,1)  =>  D = |v|^2 - 2 x.v.
// Missing per-point |x|^2 is lane-uniform (lane owns one point column of D),
// so it cannot change top-k ranking; added once in the epilogue.
//
// Hot-loop design:
//  * vpk is padded to whole 16-vert tiles (pad |v|^2 = +inf) -> the A-tile
//    load is a single unconditional 8B load, no EXEC divergence in the loop.
//  * tile loop unrolled x2 with rotated buffers so each load overlaps a full
//    process block (wait need not drain to loadcnt==0 before the WMMA).
//  * each candidate guarded by a wave-uniform ballot (v_cmp + scalar branch,
//    EXEC stays all-ones -> WMMA-legal); the branchless sorted insert runs
//    only when some lane can actually place (~10% of candidate slots).

typedef __attribute__((ext_vector_type(2))) float v2f;
typedef __attribute__((ext_vector_type(8))) float v8f;

#define WEIGHT_STD2   0.02f     // 2 * 0.1^2
#define DIS_THRESHOLD 0.2f
#define BIGF          1e30f

// Prologue: pack verts as float4 (x, y, z, |v|^2); pad tail tiles with
// |v|^2 = +inf so padded rows can never enter a top-4 list.
__global__ void pack_verts_kernel(const float* __restrict__ verts,
                                  float4* __restrict__ vpk, int V, int Vpad) {
    int v = blockIdx.x * blockDim.x + threadIdx.x;
    if (v < Vpad) {
        if (v < V) {
            float x = verts[3 * v + 0];
            float y = verts[3 * v + 1];
            float z = verts[3 * v + 2];
            vpk[v] = make_float4(x, y, z, x * x + y * y + z * z);
        } else {
            vpk[v] = make_float4(0.f, 0.f, 0.f, BIGF);
        }
    }
}

// Branchless sorted insert of (c, ci) into ascending (d0..d3, i0..i3).
#define TOP4_INSERT(cv, civ)                                        \
    do {                                                            \
        float _c = (cv); int _ci = (civ);                           \
        bool _l = _c < d3;                                          \
        d3 = _l ? _c : d3;  i3 = _l ? _ci : i3;                     \
        { bool _s = d3 < d2; float _td = d3; int _ti = i3;          \
          d3 = _s ? d2 : d3; i3 = _s ? i2 : i3;                     \
          d2 = _s ? _td : d2; i2 = _s ? _ti : i2; }                 \
        { bool _s = d2 < d1; float _td = d2; int _ti = i2;          \
          d2 = _s ? d1 : d2; i2 = _s ? i1 : i2;                     \
          d1 = _s ? _td : d1; i1 = _s ? _ti : i1; }                 \
        { bool _s = d1 < d0; float _td = d1; int _ti = i1;          \
          d1 = _s ? d0 : d1; i1 = _s ? i0 : i1;                     \
          d0 = _s ? _td : d0; i0 = _s ? _ti : i0; }                 \
    } while (0)

// Wave-uniform guard: only run the insert if ANY lane's candidate can place.
// Ballot is uniform -> scalar branch, EXEC untouched (WMMA-safe).
#define TOP4_TRY(cv, civ)                                           \
    do {                                                            \
        float _cc = (cv);                                           \
        if (__ballot(_cc < d3) != 0ull) { TOP4_INSERT(_cc, (civ)); }\
    } while (0)

__global__ __launch_bounds__(256)
void knn_lbs_kernel(const float* __restrict__ xyz,
                    const float4* __restrict__ vpk,   // padded to whole tiles
                    const float* __restrict__ Tinv,   // [V,4,4]
                    const float* __restrict__ lbs,    // [V,L]
                    float* __restrict__ out,          // [3N | N]
                    int N, int V, int L) {
    const int lane  = threadIdx.x & 31;
    const int wave  = threadIdx.x >> 5;
    const int col   = lane & 15;
    const bool hi   = lane >= 16;
    const int pbase = (blockIdx.x * (blockDim.x >> 5) + wave) * 16;
    const int p     = pbase + col;
    const int pc    = p < N ? p : N - 1;   // clamped load index

    // ---- B matrix (points side), constant over the vert loop ----
    // B is 4x16 (K x N): lane holds column N = lane&15; lo lanes K=0,1; hi K=2,3.
    float c0, c1;
    if (!hi) { c0 = xyz[3 * pc + 0]; c1 = xyz[3 * pc + 1]; }
    else     { c0 = xyz[3 * pc + 2]; c1 = 0.0f; }
    v2f bvec;
    bvec.x = -2.0f * c0;                       // -2x (lo) / -2z (hi)
    bvec.y = hi ? 1.0f : (-2.0f * c1);         // -2y (lo) / 1   (hi)
    // bring z to the lo half for the epilogue (full EXEC here)
    float zv = __shfl_xor(c0, 16, 32);

    // ---- running top-4 (ascending) of d2' = |v|^2 - 2 x.v ----
    float d0 = BIGF, d1 = BIGF, d2 = BIGF, d3 = BIGF;
    int   i0 = 0,    i1 = 0,    i2 = 0,    i3 = 0;
    v8f czero = {0.f, 0.f, 0.f, 0.f, 0.f, 0.f, 0.f, 0.f};

    const int tiles = (V + 15) >> 4;
    const int aoff  = hi ? 8 : 0;   // lo: (vx,vy)=K0,1   hi: (vz,|v|^2)=K2,3

    // Per-lane base; tile t lives at byte offset t*256 (16 verts * 16B).
    const char* abase = (const char*)vpk + (col << 4) + aoff;
    auto load_a = [&](int t) -> v2f {
        return *(const v2f*)(abase + ((size_t)(unsigned)t << 8));
    };
    auto process = [&](v2f a, int t) {
        // D[m][n] = |v_m|^2 - 2 x_n . v_m     (16x16 tile)
        v8f dmat = __builtin_amdgcn_wmma_f32_16x16x4_f32(
            false, a, false, bvec, (short)0, czero, false, false);
        // lane owns column n = col, rows m = r + 8*hi -> vert = 16*t + 8*hi + r
        const int ib = (t << 4) + aoff;
        TOP4_TRY(dmat[0], ib + 0);
        TOP4_TRY(dmat[1], ib + 1);
        TOP4_TRY(dmat[2], ib + 2);
        TOP4_TRY(dmat[3], ib + 3);
        TOP4_TRY(dmat[4], ib + 4);
        TOP4_TRY(dmat[5], ib + 5);
        TOP4_TRY(dmat[6], ib + 6);
        TOP4_TRY(dmat[7], ib + 7);
    };

    // Unrolled x2, loads issued one process-block ahead of their use.
    v2f a0 = load_a(0);
    int t = 0;
    for (; t + 2 <= tiles; t += 2) {
        v2f a1 = load_a(t + 1);
        process(a0, t);                                  // a1 in flight
        v2f a2 = load_a(t + 2 < tiles ? t + 2 : 0);      // scalar clamp
        process(a1, t + 1);                              // a2 in flight
        a0 = a2;
    }
    if (t < tiles) process(a0, t);

    // ---- merge lane p (rows 0..7) with lane p+16 (rows 8..15) ----
    // Snapshot partner values BEFORE inserting so both sides merge originals.
    float e0 = __shfl_xor(d0, 16, 32), e1 = __shfl_xor(d1, 16, 32);
    float e2 = __shfl_xor(d2, 16, 32), e3 = __shfl_xor(d3, 16, 32);
    int   j0 = __shfl_xor(i0, 16, 32), j1 = __shfl_xor(i1, 16, 32);
    int   j2 = __shfl_xor(i2, 16, 32), j3 = __shfl_xor(i3, 16, 32);
    TOP4_INSERT(e0, j0);
    TOP4_INSERT(e1, j1);
    TOP4_INSERT(e2, j2);
    TOP4_INSERT(e3, j3);

    // ---- epilogue: lo half owns the 16 points of this wave ----
    if (!hi && p < N) {
        const float xn2 = c0 * c0 + c1 * c1 + zv * zv;   // |x|^2, deferred term
        float dist0 = sqrtf(fmaxf(d0 + xn2, 1e-12f));
        float dist1 = sqrtf(fmaxf(d1 + xn2, 1e-12f));
        float dist2 = sqrtf(fmaxf(d2 + xn2, 1e-12f));
        float dist3 = sqrtf(fmaxf(d3 + xn2, 1e-12f));

        // LBS confidence gate vs nearest neighbor (conf for j=0 is exp(0)=1 > 0.9)
        const float* W0 = lbs + (size_t)i0 * L;
        const float* W1 = lbs + (size_t)i1 * L;
        const float* W2 = lbs + (size_t)i2 * L;
        const float* W3 = lbs + (size_t)i3 * L;
        float s1 = 0.f, s2s = 0.f, s3 = 0.f;
        for (int l = 0; l < L; ++l) {
            float w0l = W0[l];
            s1  += fabsf(W1[l] - w0l);
            s2s += fabsf(W2[l] - w0l);
            s3  += fabsf(W3[l] - w0l);
        }
        float conf1 = (expf(-s1  / WEIGHT_STD2) > 0.9f) ? 1.f : 0.f;
        float conf2 = (expf(-s2s / WEIGHT_STD2) > 0.9f) ? 1.f : 0.f;
        float conf3 = (expf(-s3  / WEIGHT_STD2) > 0.9f) ? 1.f : 0.f;

        float w0 = expf(-dist0);              // conf0 == 1 always
        float w1 = expf(-dist1) * conf1;
        float w2 = expf(-dist2) * conf2;
        float w3 = expf(-dist3) * conf3;
        float inv = 1.0f / (w0 + w1 + w2 + w3);
        w0 *= inv; w1 *= inv; w2 *= inv; w3 *= inv;

        float xdist = w0 * dist0 + w1 * dist1 + w2 * dist2 + w3 * dist3;
        float validf = (xdist < DIS_THRESHOLD) ? 1.0f : 0.0f;

        // Blend rows 0..2 of the 4x4 inverse transforms (row 3 never used).
        const float* T0 = Tinv + (size_t)i0 * 16;
        const float* T1 = Tinv + (size_t)i1 * 16;
        const float* T2 = Tinv + (size_t)i2 * 16;
        const float* T3 = Tinv + (size_t)i3 * 16;
        float m[12];
#pragma unroll
        for (int r = 0; r < 12; ++r)
            m[r] = w0 * T0[r] + w1 * T1[r] + w2 * T2[r] + w3 * T3[r];

        const float X = c0, Y = c1, Z = zv;
        float ox = m[0] * X + m[1] * Y + m[2]  * Z + m[3];
        float oy = m[4] * X + m[5] * Y + m[6]  * Z + m[7];
        float oz = m[8] * X + m[9] * Y + m[10] * Z + m[11];

        out[3 * p + 0] = ox;
        out[3 * p + 1] = oy;
        out[3 * p + 2] = oz;
        out[3 * N + p] = validf;
    }
}

extern "C" void kernel_launch(void* const* d_in, const int* in_sizes, int n_in,
                              void* d_out, int out_size, void* d_ws, size_t ws_size,
                              hipStream_t stream) {
    const float* xyz   = (const float*)d_in[0];   // [1,N,3]
    const float* verts = (const float*)d_in[1];   // [1,V,3]
    const float* Tinv  = (const float*)d_in[2];   // [1,V,4,4]
    const float* lbs   = (const float*)d_in[3];   // [V,L]

    const int N = in_sizes[0] / 3;
    const int V = in_sizes[1] / 3;
    const int L = in_sizes[3] / V;

    const int tiles = (V + 15) / 16;
    const int Vpad  = tiles * 16;

    // Workspace: packed verts (x,y,z,|v|^2), Vpad * 16 bytes (~110 KB here).
    float4* vpk = (float4*)d_ws;
    (void)ws_size;

    pack_verts_kernel<<<(Vpad + 255) / 256, 256, 0, stream>>>(verts, vpk, V, Vpad);

    // 16 points per wave, 8 waves (256 threads) per block.
    const int waves  = (N + 15) / 16;
    const int blocks = (waves + 7) / 8;
    knn_lbs_kernel<<<blocks, 256, 0, stream>>>(xyz, vpk, Tinv, lbs,
                                               (float*)d_out, N, V, L);
}